// spatial_pool2d_34565896798756
// MI455X (gfx1250) — compile-verified
//
#include <hip/hip_runtime.h>

// Global max-pool over spatial dims: X[B,C,H,W] -> out[B,C]
// B=16, C=256, H=W=128  =>  4096 planes of 16384 fp32 each (64 KiB/plane).
// Memory-bound: 256 MiB streamed once => NT (non-temporal) b128 loads,
// wave32 shuffle reduction, one workgroup (8 wave32s) per plane.

typedef float v4f __attribute__((ext_vector_type(4)));

#define POOL_HW     16384                      // 128*128 elements per plane
#define POOL_BLOCK  256                        // 8 wave32 waves per WG
#define POOL_V4     (POOL_HW / 4 / POOL_BLOCK) // 16 float4 loads per thread

__global__ __launch_bounds__(POOL_BLOCK) void spatial_pool2d_34565896798756_kernel(
    const float* __restrict__ X, float* __restrict__ out) {
  const int plane = blockIdx.x;                       // one (b,c) plane per WG
  const v4f* __restrict__ p =
      (const v4f*)(X + (size_t)plane * (size_t)POOL_HW);
  const int t = threadIdx.x;

  // ---- per-thread max over 64 elements via 16 NT b128 loads ----
  // Stride-POOL_BLOCK float4 access: each iteration the 256 threads cover a
  // contiguous 4 KiB span -> fully coalesced; th:TH_LOAD_NT keeps the 256 MiB
  // stream from evicting L2 (stream > 192 MB L2).
  float m = -__builtin_inff();
#pragma unroll
  for (int k = 0; k < POOL_V4; ++k) {
    v4f v = __builtin_nontemporal_load(&p[t + k * POOL_BLOCK]);
    m = fmaxf(m, fmaxf(fmaxf(v.x, v.y), fmaxf(v.z, v.w)));
  }

  // ---- wave32 butterfly reduction (CDNA5 is wave32-only) ----
#pragma unroll
  for (int off = 16; off >= 1; off >>= 1) {
    m = fmaxf(m, __shfl_xor(m, off, 32));
  }

  // ---- combine the 8 waves of this workgroup through LDS ----
  __shared__ float wave_max[POOL_BLOCK / 32];
  if ((t & 31) == 0) wave_max[t >> 5] = m;
  __syncthreads();

  if (t < POOL_BLOCK / 32) {            // lanes 0..7 of wave 0
    float w = wave_max[t];
#pragma unroll
    for (int off = 4; off >= 1; off >>= 1) {
      w = fmaxf(w, __shfl_xor(w, off, 32));   // stays within lanes 0..7
    }
    if (t == 0) out[plane] = w;
  }
}

extern "C" void kernel_launch(void* const* d_in, const int* in_sizes, int n_in,
                              void* d_out, int out_size, void* d_ws, size_t ws_size,
                              hipStream_t stream) {
  (void)n_in; (void)d_ws; (void)ws_size;
  const float* X = (const float*)d_in[0];
  float* out = (float*)d_out;

  // out_size == B*C == number of planes; in_sizes[0]/out_size == 16384 == HW.
  const int planes = out_size;
  (void)in_sizes;

  spatial_pool2d_34565896798756_kernel<<<planes, POOL_BLOCK, 0, stream>>>(X, out);
}